// GATModel_91233695302354
// MI455X (gfx1250) — compile-verified
//
#include <hip/hip_runtime.h>
#include <hip/hip_bf16.h>
#include <math.h>

typedef __attribute__((ext_vector_type(2))) float v2f;
typedef __attribute__((ext_vector_type(4))) float v4f;
typedef __attribute__((ext_vector_type(8))) float v8f;

#define NEG_SLOPE 0.01f
#define D_DIM 1447
#define H_DIM 64

__device__ __forceinline__ float lrelu(float v) { return v >= 0.f ? v : NEG_SLOPE * v; }

// ---- workspace layout (float offsets), computed in kernel_launch ----
// [0:64) v1 | [64:128) v2 | [128] c1 | [129] c2 | [130] s1max | [160:288) block maxes
// [320:320+N) s1 | [+N) s2 | [320+2N : +N*64) h (16B aligned) | [...] h2

// ---------------------------------------------------------------------------
// K0: v1 = W_tr^T a1, v2 = W_tr^T a2, c1 = b_tr.a1, c2 = b_tr.a2
// ---------------------------------------------------------------------------
__global__ void prep_kernel(const float* __restrict__ W_tr, const float* __restrict__ b_tr,
                            const float* __restrict__ a, float* __restrict__ ws) {
    int k = threadIdx.x;  // 0..63
    float v1 = 0.f, v2 = 0.f;
    for (int r = 0; r < H_DIM; ++r) {
        float w = W_tr[r * H_DIM + k];
        v1 += w * a[r];
        v2 += w * a[H_DIM + r];
    }
    ws[k] = v1;
    ws[64 + k] = v2;
    if (k == 0) {
        float c1 = 0.f, c2 = 0.f;
        for (int r = 0; r < H_DIM; ++r) {
            c1 += b_tr[r] * a[r];
            c2 += b_tr[r] * a[H_DIM + r];
        }
        ws[128] = c1;
        ws[129] = c2;
    }
}

// ---------------------------------------------------------------------------
// K1: h = lrelu(x @ W_fc^T + b_fc)   via V_WMMA_F32_16X16X4_F32
// A layout (16x4 f32): lanes 0-15 -> (M=l, K=0..1), lanes 16-31 -> (M=l, K=2..3)
// B layout (4x16 f32): lanes 0-15 -> (K=0..1, N=l), lanes 16-31 -> (K=2..3, N=l)
// ---------------------------------------------------------------------------
__global__ void fc_kernel(const float* __restrict__ x, const float* __restrict__ W_fc,
                          const float* __restrict__ b_fc, float* __restrict__ h) {
    int wave = threadIdx.x >> 5;
    int lane = threadIdx.x & 31;
    int hw = lane >> 4;
    int l = lane & 15;
    int row0 = (blockIdx.x * 8 + wave) * 16;
    const float* xrow = x + (size_t)(row0 + l) * D_DIM;

    v8f acc[4] = {};
    const int kmain = D_DIM & ~3;  // 1444
    for (int k = 0; k < kmain; k += 4) {
        int kk = k + 2 * hw;
        v2f A;
        A.x = xrow[kk];
        A.y = xrow[kk + 1];
#pragma unroll
        for (int t = 0; t < 4; ++t) {
            const float* wrow = W_fc + (size_t)(t * 16 + l) * D_DIM + kk;
            v2f B;
            B.x = wrow[0];
            B.y = wrow[1];
            acc[t] = __builtin_amdgcn_wmma_f32_16x16x4_f32(false, A, false, B, (short)0,
                                                           acc[t], false, false);
        }
    }
    {   // K tail: 1444..1446 (element 1447 is OOB -> 0)
        int kk = kmain + 2 * hw;
        v2f A;
        A.x = (kk < D_DIM) ? xrow[kk] : 0.f;
        A.y = (kk + 1 < D_DIM) ? xrow[kk + 1] : 0.f;
#pragma unroll
        for (int t = 0; t < 4; ++t) {
            const float* wrow = W_fc + (size_t)(t * 16 + l) * D_DIM;
            v2f B;
            B.x = (kk < D_DIM) ? wrow[kk] : 0.f;
            B.y = (kk + 1 < D_DIM) ? wrow[kk + 1] : 0.f;
            acc[t] = __builtin_amdgcn_wmma_f32_16x16x4_f32(false, A, false, B, (short)0,
                                                           acc[t], false, false);
        }
    }
    // D layout: lane, vgpr e -> M = e + 8*hw, N = t*16 + l
#pragma unroll
    for (int t = 0; t < 4; ++t) {
        int col = t * 16 + l;
        float bias = b_fc[col];
#pragma unroll
        for (int e = 0; e < 8; ++e) {
            int row = row0 + e + 8 * hw;
            h[(size_t)row * H_DIM + col] = lrelu(acc[t][e] + bias);
        }
    }
}

// ---------------------------------------------------------------------------
// K2: s1[n] = h[n].v1 + c1, s2[n] = h[n].v2 + c2, per-block max of s1
// ---------------------------------------------------------------------------
__global__ void score_kernel(const float* __restrict__ h, const float* __restrict__ ws,
                             float* __restrict__ s1, float* __restrict__ s2,
                             float* __restrict__ blockmax) {
    __shared__ float red[256];
    int r = blockIdx.x * 256 + threadIdx.x;
    const float* hr = h + (size_t)r * H_DIM;
    float a1 = 0.f, a2 = 0.f;
    for (int c = 0; c < H_DIM; ++c) {
        float hv = hr[c];
        a1 += hv * ws[c];        // v1
        a2 += hv * ws[64 + c];   // v2
    }
    float s1v = a1 + ws[128];
    s1[r] = s1v;
    s2[r] = a2 + ws[129];
    red[threadIdx.x] = s1v;
    __syncthreads();
    for (int s = 128; s > 0; s >>= 1) {
        if (threadIdx.x < s) red[threadIdx.x] = fmaxf(red[threadIdx.x], red[threadIdx.x + s]);
        __syncthreads();
    }
    if (threadIdx.x == 0) blockmax[blockIdx.x] = red[0];
}

__global__ void maxred_kernel(const float* __restrict__ blockmax, float* __restrict__ s1max,
                              int nb) {
    int t = threadIdx.x;
    float m = -3.4e38f;
    for (int i = t; i < nb; i += 32) m = fmaxf(m, blockmax[i]);
    for (int s = 16; s > 0; s >>= 1) m = fmaxf(m, __shfl_xor(m, s, 32));
    if (t == 0) *s1max = m;
}

// ---------------------------------------------------------------------------
// K3: fused softmax-attention + residual (flash-style; no NxN score matrix):
//   h2[i,:] = (sum_j w_ij h[j,:]) / (sum_j w_ij) + h[i,:]
//   w_ij = exp(lrelu(s2[i]+s1[j]) - m_i),  m_i = lrelu(s2[i] + max_j s1[j])
// Block = 8 waves = 128 rows; j streamed in 64-wide LDS tiles of h,
// double-buffered via CDNA5 GLOBAL_LOAD_ASYNC_TO_LDS_B128 (ASYNCcnt-tracked).
// ---------------------------------------------------------------------------
__global__ void attn_kernel(const float* __restrict__ h, const float* __restrict__ s1,
                            const float* __restrict__ s2, const float* __restrict__ s1max_p,
                            float* __restrict__ h2, int N) {
    __shared__ __align__(16) float sh[2][64 * H_DIM];  // 2 x 16 KB h tiles
    __shared__ float ss1[2][64];
    int wave = threadIdx.x >> 5;
    int lane = threadIdx.x & 31;
    int hw = lane >> 4;
    int l = lane & 15;
    int row0 = blockIdx.x * 128 + wave * 16;

    float s2i = s2[row0 + l];           // this lane's output row i = row0 + l
    float mi = lrelu(s2i + *s1max_p);   // exact row max of lrelu(scores)

    v8f acc[4] = {};
    float rsum = 0.f;
    int off = 2 * hw;  // K slots 0,1 (lanes 0-15) / 2,3 (lanes 16-31)

    // this thread's slice of a 64x64 tile: 4 x b128 (16B) transfers
    int jrow0 = threadIdx.x >> 4;        // 0..15
    int c4 = (threadIdx.x & 15) * 4;     // 0,4,..,60

    // prologue: async-stage tile 0 into buffer 0 (4 ASYNCcnt per wave)
#pragma unroll
    for (int it = 0; it < 4; ++it) {
        unsigned lds = (unsigned)(uintptr_t)&sh[0][(jrow0 + it * 16) * H_DIM + c4];
        unsigned long long ga =
            (unsigned long long)(uintptr_t)&h[(size_t)(jrow0 + it * 16) * H_DIM + c4];
        asm volatile("global_load_async_to_lds_b128 %0, %1, off" ::"v"(lds), "v"(ga)
                     : "memory");
    }
    if (threadIdx.x < 64) ss1[0][threadIdx.x] = s1[threadIdx.x];

    int ntiles = N / 64;
    for (int tile = 0; tile < ntiles; ++tile) {
        int cur = tile & 1;
        int j0n = (tile + 1) * 64;
        if (tile + 1 < ntiles) {
            // issue next tile into the other buffer, then wait for current tile
            // (async loads complete in order: <=4 outstanding => first 4 done)
#pragma unroll
            for (int it = 0; it < 4; ++it) {
                unsigned lds = (unsigned)(uintptr_t)&sh[cur ^ 1][(jrow0 + it * 16) * H_DIM + c4];
                unsigned long long ga =
                    (unsigned long long)(uintptr_t)&h[(size_t)(j0n + jrow0 + it * 16) * H_DIM + c4];
                asm volatile("global_load_async_to_lds_b128 %0, %1, off" ::"v"(lds), "v"(ga)
                             : "memory");
            }
            if (threadIdx.x < 64) ss1[cur ^ 1][threadIdx.x] = s1[j0n + threadIdx.x];
            asm volatile("s_wait_asynccnt 0x4" ::: "memory");
        } else {
            asm volatile("s_wait_asynccnt 0x0" ::: "memory");
        }
        __syncthreads();  // current tile visible to all waves

        const float* shc = sh[cur];
        const float* s1c = ss1[cur];
#pragma unroll 4
        for (int jj = 0; jj < 64; jj += 4) {
            float w0 = __expf(lrelu(s2i + s1c[jj + off]) - mi);
            float w1 = __expf(lrelu(s2i + s1c[jj + off + 1]) - mi);
            rsum += w0 + w1;
            v2f A;
            A.x = w0;
            A.y = w1;
#pragma unroll
            for (int t = 0; t < 4; ++t) {
                v2f B;
                B.x = shc[(jj + off) * H_DIM + t * 16 + l];
                B.y = shc[(jj + off + 1) * H_DIM + t * 16 + l];
                acc[t] = __builtin_amdgcn_wmma_f32_16x16x4_f32(false, A, false, B, (short)0,
                                                               acc[t], false, false);
            }
        }
        __syncthreads();  // everyone done reading buf 'cur' before it is re-filled
    }

    // rowsum for i=row0+l is split across the two half-waves (K slots 0,1 vs 2,3)
    float rtot = rsum + __shfl_xor(rsum, 16, 32);
    float inv = 1.0f / rtot;
    float inve[8];
#pragma unroll
    for (int e = 0; e < 8; ++e) inve[e] = __shfl(inv, e + 8 * hw, 32);  // row e+8*hw lives in lane e+8*hw

#pragma unroll
    for (int t = 0; t < 4; ++t) {
        int col = t * 16 + l;
#pragma unroll
        for (int e = 0; e < 8; ++e) {
            int row = row0 + e + 8 * hw;
            size_t o = (size_t)row * H_DIM + col;
            h2[o] = acc[t][e] * inve[e] + h[o];  // + residual
        }
    }
}

// ---------------------------------------------------------------------------
// K4: ans = lrelu(h2 @ W_f^T + b_f) @ W_o^T + b_o
// ---------------------------------------------------------------------------
__global__ void final_kernel(const float* __restrict__ h2, const float* __restrict__ W_f,
                             const float* __restrict__ b_f, const float* __restrict__ W_o,
                             const float* __restrict__ b_o, float* __restrict__ out) {
    int wave = threadIdx.x >> 5;
    int lane = threadIdx.x & 31;
    int hw = lane >> 4;
    int l = lane & 15;
    int row0 = (blockIdx.x * 8 + wave) * 16;
    const float* hrow = h2 + (size_t)(row0 + l) * H_DIM;

    v8f acc[4] = {};
#pragma unroll 4
    for (int k = 0; k < H_DIM; k += 4) {
        int kk = k + 2 * hw;
        v2f A = *(const v2f*)(hrow + kk);
#pragma unroll
        for (int t = 0; t < 4; ++t) {
            v2f B = *(const v2f*)(W_f + (size_t)(t * 16 + l) * H_DIM + kk);
            acc[t] = __builtin_amdgcn_wmma_f32_16x16x4_f32(false, A, false, B, (short)0,
                                                           acc[t], false, false);
        }
    }
    float part[8];
#pragma unroll
    for (int e = 0; e < 8; ++e) part[e] = 0.f;
#pragma unroll
    for (int t = 0; t < 4; ++t) {
        int col = t * 16 + l;
        float bias = b_f[col];
        float wo = W_o[col];
#pragma unroll
        for (int e = 0; e < 8; ++e) part[e] += lrelu(acc[t][e] + bias) * wo;
    }
    // reduce over the 16 columns held across lanes of each half-wave
#pragma unroll
    for (int e = 0; e < 8; ++e) {
#pragma unroll
        for (int m = 8; m >= 1; m >>= 1) part[e] += __shfl_xor(part[e], m, 16);
    }
    if (l == 0) {
        float bo = b_o[0];
#pragma unroll
        for (int e = 0; e < 8; ++e) out[row0 + e + 8 * hw] = part[e] + bo;
    }
}

// ---------------------------------------------------------------------------
extern "C" void kernel_launch(void* const* d_in, const int* in_sizes, int n_in,
                              void* d_out, int out_size, void* d_ws, size_t ws_size,
                              hipStream_t stream) {
    const float* x    = (const float*)d_in[0];
    const float* W_fc = (const float*)d_in[1];
    const float* b_fc = (const float*)d_in[2];
    const float* W_tr = (const float*)d_in[3];
    const float* b_tr = (const float*)d_in[4];
    const float* a    = (const float*)d_in[5];
    const float* W_f  = (const float*)d_in[6];
    const float* b_f  = (const float*)d_in[7];
    const float* W_o  = (const float*)d_in[8];
    const float* b_o  = (const float*)d_in[9];
    float* out = (float*)d_out;
    float* ws  = (float*)d_ws;

    int N = in_sizes[0] / D_DIM;  // 8192

    float* blockmax = ws + 160;
    float* s1max    = ws + 130;
    float* s1 = ws + 320;
    float* s2 = s1 + N;
    float* h  = s2 + N;            // offset 320+2N floats -> 16B aligned
    float* h2 = h + (size_t)N * H_DIM;

    prep_kernel<<<1, 64, 0, stream>>>(W_tr, b_tr, a, ws);
    fc_kernel<<<N / 128, 256, 0, stream>>>(x, W_fc, b_fc, h);
    score_kernel<<<N / 256, 256, 0, stream>>>(h, ws, s1, s2, blockmax);
    maxred_kernel<<<1, 32, 0, stream>>>(blockmax, s1max, N / 256);
    attn_kernel<<<N / 128, 256, 0, stream>>>(h, s1, s2, s1max, h2, N);
    final_kernel<<<N / 128, 256, 0, stream>>>(h2, W_f, b_f, W_o, b_o, out);
}